// Polynomial_54099408060720
// MI455X (gfx1250) — compile-verified
//
#include <hip/hip_runtime.h>
#include <math.h>

// out[32] = sum_d coeff_d @ row_d, degree<=3 polynomial features of x[256].
// HBM-bound GEMV (362MB f32 coeff_deg_3 streamed once). Monomials decoded on
// the fly from closed-form tetrahedral/triangular index inverses; coefficients
// streamed Global->LDS by the Tensor Data Mover (double-buffered, TENSORcnt),
// MACs by V_WMMA_F32_16X16X4_F32 with B (monomial values) replicated across
// the 16 columns -> N==0 lanes of D hold the 32 partial outputs directly.

typedef __attribute__((ext_vector_type(2))) float v2f;
typedef __attribute__((ext_vector_type(8))) float v8f;
typedef __attribute__((ext_vector_type(4))) unsigned int u32x4;
typedef __attribute__((ext_vector_type(8))) int i32x8;
typedef __attribute__((ext_vector_type(4))) int i32x4;

#if defined(__has_builtin)
#  if __has_builtin(__builtin_amdgcn_tensor_load_to_lds)
#    define HAS_TDM 1
#  endif
#endif
#ifndef HAS_TDM
#  define HAS_TDM 0
#endif

constexpr int NX   = 256;
constexpr int M3   = 2829056;             // C(258,3) = 128 * 22102
constexpr int M2   = 32896;               // C(257,2) = 128 * 257
constexpr int M1   = 256;                 //           = 128 * 2
constexpr int TILE = 128;                 // cols per tile: ALL tiles are full
constexpr int NT3  = M3 / TILE;           // 22102
constexpr int NT2  = M2 / TILE;           // 257
constexpr int NT1  = M1 / TILE;           // 2
constexpr int NTT  = NT3 + NT2 + NT1;     // 22361
constexpr int GRID = 1024;                // partials = 128KB in d_ws
constexpr int PITCH = TILE + 4;           // LDS row pitch (dwords); 132 % 64 == 4
constexpr int BUFDW = 32 * PITCH;         // 4224 dwords per buffer (16896 B)

__device__ __forceinline__ int tri(int v) { return (v * (v - 1)) >> 1; }
__device__ __forceinline__ int tet(int w) { return (w * (w - 1) * (w - 2)) / 6; }

__device__ __forceinline__ int inv_tri(int u) {   // smallest v: tri(v) >= u
  int v = (int)sqrtf(2.0f * (float)u) + 1;
  while (v > 2 && tri(v - 1) >= u) --v;
  while (tri(v) < u) ++v;
  return v;
}
__device__ __forceinline__ int inv_tet(int u) {   // smallest w: tet(w) >= u
  int w = (int)cbrtf(6.0f * (float)u) + 1;
  while (w > 3 && tet(w - 1) >= u) --w;
  while (tet(w) < u) ++w;
  return w;
}

struct TileInfo { const float* cp; int M; int k0; int deg; };

__device__ __forceinline__ TileInfo classify(int tile, const float* c1,
                                             const float* c2, const float* c3) {
  TileInfo t;
  if (tile < NT3)            { t.cp = c3; t.M = M3; t.k0 = tile * TILE;               t.deg = 3; }
  else if (tile < NT3 + NT2) { t.cp = c2; t.M = M2; t.k0 = (tile - NT3) * TILE;       t.deg = 2; }
  else                       { t.cp = c1; t.M = M1; t.k0 = (tile - NT3 - NT2) * TILE; t.deg = 1; }
  return t;
}

#if HAS_TDM
// Issue one TDM 2D tile load: 32 rows x 128 cols of f32, row stride M,
// into LDS with 4-dword padding after every 128 dwords (pitch 132 -> the
// 16-row WMMA A-operand ds_load_b64 pattern is bank-conflict-free).
// D# packing per CDNA5 ISA §8.3 (group0) / §8.4 (group1); groups 2/3 and the
// trailing i32x8 are unused for a 2D tile (tile_dim2 == 0) and passed as zero.
__device__ __forceinline__ void tdm_issue(const float* gsrc, int M, float* ldsDst) {
  typedef __attribute__((address_space(3))) float lds_f;
  unsigned lds_off = (unsigned)(size_t)(lds_f*)ldsDst;          // LDS byte offset
  unsigned long long ga = (unsigned long long)(size_t)gsrc;     // 57-bit global addr
  u32x4 g0;
  g0[0] = 1u;                                   // count=1, user descriptor
  g0[1] = lds_off;                              // lds_addr
  g0[2] = (unsigned)ga;                         // global_addr[31:0]
  g0[3] = (unsigned)(ga >> 32) | (2u << 30);    // global_addr[56:32] | type=2
  i32x8 g1;
  g1[0] = (int)(0x00020000u                     // data_size = 2 (4 bytes)
              | 0x00100000u                     // pad_enable
              | (6u << 22)                      // pad_interval: 128 dwords
              | (3u << 25));                    // pad_amount:   4 dwords
  g1[1] = (int)(((unsigned)M & 0xFFFFu) << 16); // tensor_dim0[15:0]
  g1[2] = (int)(((unsigned)M >> 16) | (32u << 16)); // tensor_dim0[31:16] | tensor_dim1=32
  g1[3] = (int)((unsigned)TILE << 16);          // tile_dim0 = 128
  g1[4] = 32;                                   // tile_dim1 = 32 (tile_dim2=0 -> 2D)
  g1[5] = M;                                    // tensor_dim0_stride[31:0]
  g1[6] = 0;                                    // stride0[47:32] | stride1 lo
  g1[7] = 0;
  i32x4 z4 = {};                                // groups 2/3 unused (2D)
  i32x8 z8 = {};                                // trailing group (unused)
  __builtin_amdgcn_tensor_load_to_lds(g0, g1, z4, z4, z8, 0);
}
#else
// Fallback: cooperative coalesced copy into the same padded LDS layout.
__device__ __forceinline__ void coop_copy(const float* gsrc, int M, float* dst) {
  const int t  = threadIdx.x;        // 256 threads, 16 floats each
  const int r  = t >> 3;             // row 0..31
  const int c0 = (t & 7) << 4;       // col 0,16,...,112
  const float* s = gsrc + (size_t)r * M + c0;
  float* d = dst + r * PITCH + c0;
  #pragma unroll
  for (int e = 0; e < 16; e += 4)
    *(float4*)(d + e) = *(const float4*)(s + e);
}
#endif

__global__ __launch_bounds__(256) void poly_main(
    const float* __restrict__ x,
    const float* __restrict__ c1,
    const float* __restrict__ c2,
    const float* __restrict__ c3,
    float* __restrict__ partials)
{
  __shared__ float lds_A[2][BUFDW];     // double-buffered coeff tiles (33.8 KB)
  __shared__ float lds_x[NX];
  __shared__ float lds_r[TILE];
  __shared__ float lds_part[8][32];

  lds_x[threadIdx.x] = x[threadIdx.x];

  const int lane = threadIdx.x & 31;
  const int wv   = threadIdx.x >> 5;
  const int half = lane >> 4;          // 0: K {0,1}, 1: K {2,3}
  const int row  = lane & 15;          // A-matrix output row

  v8f acc0 = {};  // outputs 0..15
  v8f acc1 = {};  // outputs 16..31

  // ---- prologue: prefetch first tile into buffer 0 ----
  {
    TileInfo t0 = classify(blockIdx.x, c1, c2, c3);
#if HAS_TDM
    if (wv == 0) tdm_issue(t0.cp + t0.k0, t0.M, &lds_A[0][0]);
#else
    coop_copy(t0.cp + t0.k0, t0.M, &lds_A[0][0]);
#endif
  }
  __syncthreads();   // lds_x visible (and fallback buf0 ready)

  int n = 0;
  for (int tile = blockIdx.x; tile < NTT; tile += gridDim.x, ++n) {
    const int bsel = n & 1;
    const int next = tile + gridDim.x;
    const bool hasNext = next < NTT;

    // 1) start streaming the NEXT tile into the other buffer
    if (hasNext) {
      TileInfo tn = classify(next, c1, c2, c3);
#if HAS_TDM
      if (wv == 0) tdm_issue(tn.cp + tn.k0, tn.M, &lds_A[bsel ^ 1][0]);
#else
      coop_copy(tn.cp + tn.k0, tn.M, &lds_A[bsel ^ 1][0]);
#endif
    }

    // 2) decode this tile's 128 monomial values (overlaps the TDM transfer)
    TileInfo ti = classify(tile, c1, c2, c3);
    if (threadIdx.x < TILE) {
      const int m = ti.k0 + threadIdx.x;
      float rv;
      if (ti.deg == 3) {
        int u  = M3 - m;
        int w  = inv_tet(u);                  // 3..258
        int i  = 258 - w;
        int m2 = m - (M3 - tet(w));
        int u2 = tri(w - 1) - m2;
        int v  = inv_tri(u2);                 // 2..257
        int j  = 257 - v;
        int k  = j + (m2 - (tri(w - 1) - tri(v)));
        rv = lds_x[i] * lds_x[j] * lds_x[k];
      } else if (ti.deg == 2) {
        int u = M2 - m;
        int v = inv_tri(u);
        int i = 257 - v;
        int j = i + (m - (M2 - tri(v)));
        rv = lds_x[i] * lds_x[j];
      } else {
        rv = lds_x[m];
      }
      lds_r[threadIdx.x] = rv;
    }

    // 3) make sure THIS tile's TDM transfer has landed, then sync the block
#if HAS_TDM
    if (wv == 0) {
      if (hasNext) __builtin_amdgcn_s_wait_tensorcnt(1);  // oldest (this tile) done
      else         __builtin_amdgcn_s_wait_tensorcnt(0);
    }
#endif
    __syncthreads();

    // 4) WMMA: wave wv owns cols [wv*16, wv*16+16) of the tile
    const float* bufp = &lds_A[bsel][0];
    #pragma unroll
    for (int s = 0; s < 4; ++s) {
      const int cb = (wv << 4) + (s << 2) + (half << 1);  // even
      v2f a0 = *(const v2f*)(bufp + row * PITCH + cb);
      v2f a1 = *(const v2f*)(bufp + (row + 16) * PITCH + cb);
      v2f b;
      b.x = lds_r[cb];
      b.y = lds_r[cb + 1];
      acc0 = __builtin_amdgcn_wmma_f32_16x16x4_f32(false, a0, false, b,
                                                   (short)0, acc0, false, false);
      acc1 = __builtin_amdgcn_wmma_f32_16x16x4_f32(false, a1, false, b,
                                                   (short)0, acc1, false, false);
    }
    __syncthreads();   // all reads of buf[bsel]/lds_r done before they are reused
  }

  // ---- per-block reduction: N==0 lanes (0,16) hold the column sums ----
  if ((lane & 15) == 0) {
    const int mb = half * 8;
    #pragma unroll
    for (int v = 0; v < 8; ++v) {
      lds_part[wv][mb + v]      = acc0[v];
      lds_part[wv][16 + mb + v] = acc1[v];
    }
  }
  __syncthreads();
  if (threadIdx.x < 32) {
    float s = 0.0f;
    #pragma unroll
    for (int w = 0; w < 8; ++w) s += lds_part[w][threadIdx.x];
    partials[(size_t)blockIdx.x * 32 + threadIdx.x] = s;
  }
}

__global__ __launch_bounds__(256) void poly_final(
    const float* __restrict__ c0,
    const float* __restrict__ partials,
    float* __restrict__ out, int nblocks)
{
  __shared__ float red[256];
  const int o  = threadIdx.x & 31;
  const int sl = threadIdx.x >> 5;
  float s = 0.0f;
  for (int b = sl; b < nblocks; b += 8)
    s += partials[(size_t)b * 32 + o];
  red[threadIdx.x] = s;
  __syncthreads();
  if (threadIdx.x < 32) {
    float t = c0[o];
    #pragma unroll
    for (int w = 0; w < 8; ++w) t += red[w * 32 + o];
    out[o] = t;
  }
}

extern "C" void kernel_launch(void* const* d_in, const int* in_sizes, int n_in,
                              void* d_out, int out_size, void* d_ws, size_t ws_size,
                              hipStream_t stream) {
  const float* x  = (const float*)d_in[0];   // (256,)
  const float* c0 = (const float*)d_in[1];   // (32, 1)
  const float* c1 = (const float*)d_in[2];   // (32, 256)
  const float* c2 = (const float*)d_in[3];   // (32, 32896)
  const float* c3 = (const float*)d_in[4];   // (32, 2829056)
  float* out      = (float*)d_out;           // (32,)
  float* partials = (float*)d_ws;            // GRID*32 floats = 128 KB

  poly_main<<<GRID, 256, 0, stream>>>(x, c1, c2, c3, partials);
  poly_final<<<1, 256, 0, stream>>>(c0, partials, out, GRID);
}